// GAT_1202590843050
// MI455X (gfx1250) — compile-verified
//
#include <hip/hip_runtime.h>
#include <hip/hip_bf16.h>

// ---------------------------------------------------------------------------
// GAT (3 layers, 4 heads, concat=False) for gfx1250 / MI455X.
// Dense projections: V_WMMA_F32_16X16X4_F32, B fragments staged through LDS
// (transposed, padded) so the inner loop feeds WMMA from ds_load_b64 instead
// of serialized per-lane global loads. Edge softmax/aggregation use L2 float
// atomics (whole graph fits in the 192MB L2).
// ---------------------------------------------------------------------------

#define NN      20000      // nodes
#define E0      320000     // raw edges
#define ET      340000     // edges + self loops
#define DIN     128
#define NC      512        // HEADS * DOUT = 4*128
#define DOUT    128
#define NEG_SLOPE 0.2f
#define BN_EPS  1e-5f

#define CG      64         // columns per LDS slab
#define LDSK    130        // padded per-column stride (floats), keeps 8B align
#define NBLK    40         // blocks per column group (persistent-ish)

typedef float v2f __attribute__((ext_vector_type(2)));
typedef float v8f __attribute__((ext_vector_type(8)));

#if __has_builtin(__builtin_amdgcn_wmma_f32_16x16x4_f32)
#define HAVE_WMMA_F32 1
#endif

// --------------------------- GEMM: H = X @ W -------------------------------
// X: [M, K] row-major, W: [K, Nc] row-major, H: [M, Nc].
// Block = 8 waves. Block owns one 64-column group: stages W[:,cg*64..+64)
// transposed into LDS once, then waves grid-stride over 16-row M-tiles.
// Each wave: 16 rows x 64 cols = 4 WMMA accumulators sharing one A fragment.
#ifdef HAVE_WMMA_F32
__global__ void __launch_bounds__(256)
k_gemm_wmma(const float* __restrict__ X, const float* __restrict__ W,
            float* __restrict__ H, int M, int K, int Nc) {
  __shared__ float ldsW[CG * LDSK];            // 33.3 KB

  const int groups = Nc / CG;                  // 8
  const int cg = blockIdx.x % groups;          // column group
  const int bm = blockIdx.x / groups;          // block index within group
  const int nb = gridDim.x / groups;           // blocks per group
  const int lane = threadIdx.x & 31;
  const int wv   = threadIdx.x >> 5;           // wave id in block (uniform)

  // ---- stage W slab, transposed: ldsW[c*LDSK + k] = W[k][cg*CG + c] ----
  for (int t = threadIdx.x; t < K * CG; t += 256) {
    int k = t / CG, c = t % CG;                // consecutive t -> consecutive c
    ldsW[c * LDSK + k] = W[(size_t)k * Nc + cg * CG + c];
  }
  __syncthreads();

  const int kh = (lane >> 4) << 1;             // lanes 0-15 -> K{0,1}; 16-31 -> K{2,3}
  const float* bp = &ldsW[(lane & 15) * LDSK + kh];
  const int mtiles = M >> 4;

  for (int mt = bm * 8 + wv; mt < mtiles; mt += nb * 8) {   // wave-uniform
    const int row = mt * 16 + (lane & 15);
    const float* xr = X + (size_t)row * K;

    v8f c0 = {0,0,0,0,0,0,0,0};
    v8f c1 = {0,0,0,0,0,0,0,0};
    v8f c2 = {0,0,0,0,0,0,0,0};
    v8f c3 = {0,0,0,0,0,0,0,0};

#pragma unroll 8
    for (int k0 = 0; k0 < DIN; k0 += 4) {
      v2f a;
      a.x = xr[k0 + kh];
      a.y = xr[k0 + kh + 1];
      v2f b0 = *(const v2f*)(bp + 0 * 16 * LDSK + k0);   // ds_load_b64
      v2f b1 = *(const v2f*)(bp + 1 * 16 * LDSK + k0);
      v2f b2 = *(const v2f*)(bp + 2 * 16 * LDSK + k0);
      v2f b3 = *(const v2f*)(bp + 3 * 16 * LDSK + k0);
      c0 = __builtin_amdgcn_wmma_f32_16x16x4_f32(false, a, false, b0, (short)0, c0, false, false);
      c1 = __builtin_amdgcn_wmma_f32_16x16x4_f32(false, a, false, b1, (short)0, c1, false, false);
      c2 = __builtin_amdgcn_wmma_f32_16x16x4_f32(false, a, false, b2, (short)0, c2, false, false);
      c3 = __builtin_amdgcn_wmma_f32_16x16x4_f32(false, a, false, b3, (short)0, c3, false, false);
    }
    // C/D layout: VGPR i -> M = i + 8*(lane>=16), N = lane&15
    const int rbase = mt * 16 + ((lane >> 4) << 3);
    const int colBase = cg * CG + (lane & 15);
#pragma unroll
    for (int i = 0; i < 8; ++i) {
      float* out = H + (size_t)(rbase + i) * Nc + colBase;
      out[0]  = c0[i];
      out[16] = c1[i];
      out[32] = c2[i];
      out[48] = c3[i];
    }
  }
}
#else
__global__ void __launch_bounds__(256)
k_gemm_scalar(const float* __restrict__ X, const float* __restrict__ W,
              float* __restrict__ H, int M, int K, int Nc) {
  int t = blockIdx.x * blockDim.x + threadIdx.x;
  if (t >= M * Nc) return;
  int row = t / Nc, col = t % Nc;
  float acc = 0.f;
  for (int k = 0; k < K; ++k) acc = fmaf(X[(size_t)row * K + k], W[(size_t)k * Nc + col], acc);
  H[t] = acc;
}
#endif

// ----------------------- per-layer init (m, z, acc) ------------------------
__global__ void k_init(float* __restrict__ m, float* __restrict__ z,
                       float* __restrict__ acc, int n) {
  int t = blockIdx.x * blockDim.x + threadIdx.x;
  if (t < n * 4) { m[t] = -3.402823466e38f; z[t] = 0.f; }
  if (t < n * 128) acc[t] = 0.f;
}

// --------------------- alpha_s / alpha_d projections -----------------------
__global__ void k_alpha(const float* __restrict__ H, const float* __restrict__ asrc,
                        const float* __restrict__ adst, float* __restrict__ als,
                        float* __restrict__ ald, int n) {
  int t = blockIdx.x * blockDim.x + threadIdx.x;   // t = node*4 + head
  if (t >= n * 4) return;
  int node = t >> 2, h = t & 3;
  const float* hp = H + (size_t)node * NC + h * DOUT;
  const float* as = asrc + h * DOUT;
  const float* ad = adst + h * DOUT;
  float s = 0.f, d = 0.f;
#pragma unroll 4
  for (int c = 0; c < DOUT; ++c) { float v = hp[c]; s = fmaf(v, as[c], s); d = fmaf(v, ad[c], d); }
  als[t] = s; ald[t] = d;
}

// ------------------------- float atomic max trick --------------------------
// Valid given init = -FLT_MAX: signed-int max for v>=0, unsigned min for v<0.
__device__ __forceinline__ void atomicMaxFloat(float* addr, float v) {
  if (v >= 0.f) atomicMax((int*)addr, __float_as_int(v));
  else          atomicMin((unsigned int*)addr, __float_as_uint(v));
}

// ------------------ leaky-relu logits + segment max ------------------------
__global__ void k_logit_max(const int* __restrict__ ei, const float* __restrict__ als,
                            const float* __restrict__ ald, float* __restrict__ logits,
                            float* __restrict__ m) {
  int t = blockIdx.x * blockDim.x + threadIdx.x;   // t = edge*4 + head
  if (t >= ET * 4) return;
  int e = t >> 2, h = t & 3;
  int s = (e < E0) ? ei[e] : (e - E0);
  int d = (e < E0) ? ei[E0 + e] : (e - E0);
  float x = als[s * 4 + h] + ald[d * 4 + h];
  float lg = (x > 0.f) ? x : NEG_SLOPE * x;
  logits[t] = lg;
  atomicMaxFloat(&m[d * 4 + h], lg);
}

// ----------------------- exp(logit - max) + segment sum --------------------
__global__ void k_expsum(const int* __restrict__ ei, float* __restrict__ logits,
                         const float* __restrict__ m, float* __restrict__ z) {
  int t = blockIdx.x * blockDim.x + threadIdx.x;
  if (t >= ET * 4) return;
  int e = t >> 2, h = t & 3;
  int d = (e < E0) ? ei[E0 + e] : (e - E0);
  float ev = expf(logits[t] - m[d * 4 + h]);
  logits[t] = ev;                                   // reuse buffer: now holds e
  atomicAdd(&z[d * 4 + h], ev);
}

// -------- scatter-aggregate: acc[dst,c] += 1/4 * sum_h alpha_h * H[src,h,c]
// one wave per edge; lane handles 4 channels (32 lanes * 4 = 128)
__global__ void __launch_bounds__(256)
k_aggregate(const int* __restrict__ ei, const float* __restrict__ H,
            const float* __restrict__ eexp, const float* __restrict__ z,
            float* __restrict__ acc) {
  int wave = (blockIdx.x * blockDim.x + threadIdx.x) >> 5;
  int lane = threadIdx.x & 31;
  if (wave >= ET) return;
  int s, d;
  if (wave < E0) { s = ei[wave]; d = ei[E0 + wave]; } else { s = d = wave - E0; }
  float w0 = 0.25f * eexp[wave * 4 + 0] / (z[d * 4 + 0] + 1e-16f);
  float w1 = 0.25f * eexp[wave * 4 + 1] / (z[d * 4 + 1] + 1e-16f);
  float w2 = 0.25f * eexp[wave * 4 + 2] / (z[d * 4 + 2] + 1e-16f);
  float w3 = 0.25f * eexp[wave * 4 + 3] / (z[d * 4 + 3] + 1e-16f);
  int c = lane << 2;
  const float* hp = H + (size_t)s * NC;
  float4 v0 = *(const float4*)(hp + 0 * DOUT + c);
  float4 v1 = *(const float4*)(hp + 1 * DOUT + c);
  float4 v2 = *(const float4*)(hp + 2 * DOUT + c);
  float4 v3 = *(const float4*)(hp + 3 * DOUT + c);
  float r0 = w0 * v0.x + w1 * v1.x + w2 * v2.x + w3 * v3.x;
  float r1 = w0 * v0.y + w1 * v1.y + w2 * v2.y + w3 * v3.y;
  float r2 = w0 * v0.z + w1 * v1.z + w2 * v2.z + w3 * v3.z;
  float r3 = w0 * v0.w + w1 * v1.w + w2 * v2.w + w3 * v3.w;
  float* ap = acc + (size_t)d * DOUT + c;
  atomicAdd(ap + 0, r0);
  atomicAdd(ap + 1, r1);
  atomicAdd(ap + 2, r2);
  atomicAdd(ap + 3, r3);
}

// ---------------------- BN column stats (mean / var) -----------------------
__global__ void __launch_bounds__(256)
k_colstats(const float* __restrict__ x, float* __restrict__ mu,
           float* __restrict__ var, int n) {
  __shared__ float sh[256], sh2[256];
  int c = blockIdx.x;                 // one block per column (128 blocks)
  float s = 0.f, s2 = 0.f;
  for (int r = threadIdx.x; r < n; r += 256) {
    float v = x[(size_t)r * 128 + c];
    s += v; s2 += v * v;
  }
  sh[threadIdx.x] = s; sh2[threadIdx.x] = s2;
  __syncthreads();
  for (int st = 128; st > 0; st >>= 1) {
    if (threadIdx.x < st) { sh[threadIdx.x] += sh[threadIdx.x + st]; sh2[threadIdx.x] += sh2[threadIdx.x + st]; }
    __syncthreads();
  }
  if (threadIdx.x == 0) {
    float mean = sh[0] / n;
    mu[c] = mean;
    var[c] = sh2[0] / n - mean * mean;
  }
}

// ---------------------------- BN + ReLU ------------------------------------
// Per-channel bias b is omitted on BN layers: BN subtracts the column mean,
// so a constant column shift cancels exactly.
__global__ void k_bnrelu(const float* __restrict__ acc, const float* __restrict__ mu,
                         const float* __restrict__ var, const float* __restrict__ gamma,
                         const float* __restrict__ beta, float* __restrict__ y, int total) {
  int t = blockIdx.x * blockDim.x + threadIdx.x;
  if (t >= total) return;
  int c = t & 127;
  float v = (acc[t] - mu[c]) * rsqrtf(var[c] + BN_EPS) * gamma[c] + beta[c];
  y[t] = (v > 0.f) ? v : 0.f;
}

// ------------------------- final layer: + bias -----------------------------
__global__ void k_bias(const float* __restrict__ acc, const float* __restrict__ b,
                       float* __restrict__ y, int total) {
  int t = blockIdx.x * blockDim.x + threadIdx.x;
  if (t >= total) return;
  y[t] = acc[t] + b[t & 127];
}

// ---------------------------------------------------------------------------
extern "C" void kernel_launch(void* const* d_in, const int* in_sizes, int n_in,
                              void* d_out, int out_size, void* d_ws, size_t ws_size,
                              hipStream_t stream) {
  (void)in_sizes; (void)n_in; (void)out_size; (void)ws_size;
  // setup_inputs() dict order:
  // 0:x 1:edge_index 2:edge_attr 3:W0 4:asrc0 5:adst0 6:b0 7:W1 8:asrc1 9:adst1
  // 10:b1 11:W2 12:asrc2 13:adst2 14:b2 15:gamma0 16:beta0 17:gamma1 18:beta1
  const float* x     = (const float*)d_in[0];
  const int*   ei    = (const int*)d_in[1];
  const float* eattr = (const float*)d_in[2];
  const float* W[3]  = {(const float*)d_in[3],  (const float*)d_in[7],  (const float*)d_in[11]};
  const float* As[3] = {(const float*)d_in[4],  (const float*)d_in[8],  (const float*)d_in[12]};
  const float* Ad[3] = {(const float*)d_in[5],  (const float*)d_in[9],  (const float*)d_in[13]};
  const float* Bv[3] = {(const float*)d_in[6],  (const float*)d_in[10], (const float*)d_in[14]};
  const float* G[2]  = {(const float*)d_in[15], (const float*)d_in[17]};
  const float* Be[2] = {(const float*)d_in[16], (const float*)d_in[18]};

  // workspace layout (floats, 256B-aligned chunks)
  float* ws = (float*)d_ws;
  size_t off = 0;
  auto alloc = [&](size_t nf) { float* p = ws + off; off += (nf + 63) & ~(size_t)63; return p; };
  float* H    = alloc((size_t)NN * NC);     // 10.24M f : projections
  float* als  = alloc((size_t)NN * 4);
  float* ald  = alloc((size_t)NN * 4);
  float* mbuf = alloc((size_t)NN * 4);
  float* zbuf = alloc((size_t)NN * 4);
  float* ebuf = alloc((size_t)ET * 4);      // logits -> exp values
  float* acc  = alloc((size_t)NN * 128);    // aggregated layer output
  float* t0   = alloc((size_t)NN * 128);    // inter-layer activations
  float* mu   = alloc(128);
  float* var  = alloc(128);

  const int blk = 256;
  const int gInit  = (NN * 128 + blk - 1) / blk;
  const int gAlpha = (NN * 4 + blk - 1) / blk;
  const int gEdge  = (ET * 4 + blk - 1) / blk;
  const int gAggr  = (ET * 32 + blk - 1) / blk;      // one wave per edge
  const int gElem  = (NN * 128 + blk - 1) / blk;
#ifdef HAVE_WMMA_F32
  const int gGemm  = (NC / CG) * NBLK;               // 8 groups x 40 blocks
#else
  const int gGemm  = (NN * NC + blk - 1) / blk;
#endif

  for (int L = 0; L < 3; ++L) {
    const float* xin = (L == 0) ? x : t0;
    k_init<<<gInit, blk, 0, stream>>>(mbuf, zbuf, acc, NN);
#ifdef HAVE_WMMA_F32
    k_gemm_wmma<<<gGemm, blk, 0, stream>>>(xin, W[L], H, NN, DIN, NC);
#else
    k_gemm_scalar<<<gGemm, blk, 0, stream>>>(xin, W[L], H, NN, DIN, NC);
#endif
    k_alpha<<<gAlpha, blk, 0, stream>>>(H, As[L], Ad[L], als, ald, NN);
    k_logit_max<<<gEdge, blk, 0, stream>>>(ei, als, ald, ebuf, mbuf);
    k_expsum<<<gEdge, blk, 0, stream>>>(ei, ebuf, mbuf, zbuf);
    k_aggregate<<<gAggr, blk, 0, stream>>>(ei, H, ebuf, zbuf, acc);
    if (L < 2) {
      k_colstats<<<128, blk, 0, stream>>>(acc, mu, var, NN);
      k_bnrelu<<<gElem, blk, 0, stream>>>(acc, mu, var, G[L], Be[L], t0, NN * 128);
    } else {
      k_bias<<<gElem, blk, 0, stream>>>(acc, Bv[2], (float*)d_out, NN * 128);
    }
  }
  // second tuple output: edge_attr passthrough
  hipMemcpyAsync((float*)d_out + (size_t)NN * 128, eattr,
                 (size_t)E0 * 16 * sizeof(float), hipMemcpyDeviceToDevice, stream);
}